// GAT_6828998000901
// MI455X (gfx1250) — compile-verified
//
#include <hip/hip_runtime.h>
#include <hip/hip_bf16.h>
#include <math.h>

typedef float v2f __attribute__((ext_vector_type(2)));
typedef float v8f __attribute__((ext_vector_type(8)));

#define N_NODES 50000
#define N_EDGES 800000
#define E2      (N_EDGES + N_NODES)   // edges + self loops
#define F_INDIM 128
#define HID     128
#define EDIM    32
#define NEG_SLOPE 0.2f

// ---------------------------------------------------------------------------
// H = X @ W  ;  X:[M,K] row-major, W:[K,HID] row-major, H:[M,HID]
// One block = 16 rows; 8 waves, wave w computes cols [16w,16w+16) with
// V_WMMA_F32_16X16X4_F32 (full fp32 matrix path).
// A 16x4 layout: VGPR0 lanes0-15 K=0 / lanes16-31 K=2; VGPR1 K=1 / K=3.
// B 4x16 layout: VGPR0 lanes0-15 row K=0 / lanes16-31 row K=2; VGPR1 K=1/K=3.
// C/D 16x16:     VGPR i lanes0-15 M=i / lanes16-31 M=8+i.
// ---------------------------------------------------------------------------
__global__ __launch_bounds__(256) void gemm_wmma_f32(
    const float* __restrict__ X, const float* __restrict__ W,
    float* __restrict__ H, int K)
{
    const int lane = threadIdx.x & 31;
    const int wave = threadIdx.x >> 5;
    const int row0 = blockIdx.x * 16;
    const int n0   = wave * 16;
    const int ml   = lane & 15;
    const int khi  = (lane >> 4) << 1;          // 0 or 2

    v8f c = {};
    const float* xrow = X + (size_t)(row0 + ml) * K;
    for (int kk = 0; kk < K; kk += 4) {
        v2f a, b;
        a.x = xrow[kk + khi];
        a.y = xrow[kk + khi + 1];
        b.x = W[(size_t)(kk + khi)     * HID + n0 + ml];
        b.y = W[(size_t)(kk + khi + 1) * HID + n0 + ml];
        c = __builtin_amdgcn_wmma_f32_16x16x4_f32(
                false, a, false, b, (short)0, c, false, false);
    }
    const int rbase = row0 + ((lane >> 4) << 3);
#pragma unroll
    for (int i = 0; i < 8; ++i)
        H[(size_t)(rbase + i) * HID + n0 + ml] = c[i];
}

// ---------------------------------------------------------------------------
// Per-node: alpha = h . a_s, beta = h . a_d (one wave / node), plus zero the
// per-node softmax state and the aggregation output buffer.
// ---------------------------------------------------------------------------
__global__ __launch_bounds__(256) void node_prep(
    const float* __restrict__ H,
    const float* __restrict__ a_s, const float* __restrict__ a_d,
    float* __restrict__ alpha, float* __restrict__ beta,
    unsigned* __restrict__ mkeys, float* __restrict__ ssum,
    float* __restrict__ aggout)
{
    const int wid  = (int)((blockIdx.x * blockDim.x + threadIdx.x) >> 5);
    const int lane = threadIdx.x & 31;
    if (wid >= N_NODES) return;

    float4 h  = ((const float4*)(H + (size_t)wid * HID))[lane];
    float4 s4 = ((const float4*)a_s)[lane];
    float4 d4 = ((const float4*)a_d)[lane];
    float pa = h.x*s4.x + h.y*s4.y + h.z*s4.z + h.w*s4.w;
    float pb = h.x*d4.x + h.y*d4.y + h.z*d4.z + h.w*d4.w;
#pragma unroll
    for (int off = 16; off; off >>= 1) {
        pa += __shfl_xor(pa, off, 32);
        pb += __shfl_xor(pb, off, 32);
    }
    if (lane == 0) {
        alpha[wid] = pa;
        beta[wid]  = pb;
        mkeys[wid] = 0u;      // encoded "very small" (every node has self loop)
        ssum[wid]  = 0.f;
    }
    ((float4*)(aggout + (size_t)wid * HID))[lane] = make_float4(0.f,0.f,0.f,0.f);
}

// ---------------------------------------------------------------------------
// Sum of edge_attr over all real edges -> msum[0..31] (divide by E later).
// ---------------------------------------------------------------------------
__global__ void zero32(float* __restrict__ p)
{
    if (threadIdx.x < EDIM) p[threadIdx.x] = 0.f;
}

__global__ __launch_bounds__(256) void mean_ea_sum(
    const float* __restrict__ ea, float* __restrict__ msum)
{
    __shared__ float acc[8 * EDIM];
    const int lane = threadIdx.x & 31;     // == dim (EDIM == 32)
    const int wv   = threadIdx.x >> 5;     // 0..7
    float partial = 0.f;
    for (long e = (long)blockIdx.x * 8 + wv; e < N_EDGES; e += (long)gridDim.x * 8)
        partial += ea[e * EDIM + lane];
    acc[wv * EDIM + lane] = partial;
    __syncthreads();
    if (wv == 0) {
        float s = 0.f;
#pragma unroll
        for (int i = 0; i < 8; ++i) s += acc[i * EDIM + lane];
        atomicAdd(&msum[lane], s);
    }
}

// ---------------------------------------------------------------------------
// weae = We @ a_e  (32 values);  sloop = dot(mean_ea, weae).  One tiny block.
// ---------------------------------------------------------------------------
__global__ void prep_edge_proj(
    const float* __restrict__ We, const float* __restrict__ a_e,
    const float* __restrict__ msum, float* __restrict__ weae,
    float* __restrict__ sloop)
{
    __shared__ float tmp[EDIM];
    const int d = threadIdx.x;            // 0..31
    float acc = 0.f;
    for (int c = 0; c < HID; ++c) acc += We[d * HID + c] * a_e[c];
    weae[d] = acc;
    tmp[d] = acc * (msum[d] * (1.0f / (float)N_EDGES));
    __syncthreads();
    if (d == 0) {
        float s = 0.f;
        for (int i = 0; i < EDIM; ++i) s += tmp[i];
        *sloop = s;
    }
}

// ---------------------------------------------------------------------------
// Pass 1: edge logits (leaky relu) + segment max via encoded-uint atomicMax.
// Handles real edges [0,E) and self loops [E,E2).
// ---------------------------------------------------------------------------
__device__ __forceinline__ unsigned enc_f32(float f) {
    unsigned u = __float_as_uint(f);
    return (u & 0x80000000u) ? ~u : (u | 0x80000000u);
}
__device__ __forceinline__ float dec_f32(unsigned k) {
    return (k & 0x80000000u) ? __uint_as_float(k ^ 0x80000000u)
                             : __uint_as_float(~k);
}

__global__ __launch_bounds__(256) void edge_logit(
    const int* __restrict__ src, const int* __restrict__ dst,
    const float* __restrict__ ea,
    const float* __restrict__ alpha, const float* __restrict__ beta,
    const float* __restrict__ weae, const float* __restrict__ sloop,
    float* __restrict__ logit, unsigned* __restrict__ mkeys)
{
    const int e = (int)(blockIdx.x * blockDim.x + threadIdx.x);
    if (e >= E2) return;
    int s, d;
    float l;
    if (e < N_EDGES) {
        s = src[e]; d = dst[e];
        const float4* row = (const float4*)(ea + (size_t)e * EDIM);
        const float4* w4  = (const float4*)weae;
        float acc = 0.f;
#pragma unroll
        for (int i = 0; i < EDIM / 4; ++i) {
            float4 r = row[i], w = w4[i];
            acc += r.x*w.x + r.y*w.y + r.z*w.z + r.w*w.w;
        }
        l = alpha[s] + beta[d] + acc;
    } else {
        s = d = e - N_EDGES;
        l = alpha[s] + beta[d] + *sloop;
    }
    l = (l > 0.f) ? l : NEG_SLOPE * l;
    logit[e] = l;
    atomicMax(&mkeys[d], enc_f32(l));
}

// ---------------------------------------------------------------------------
// Pass 2: ex = exp(logit - m[dst]); logit <- ex (in place); ssum[dst] += ex.
// ---------------------------------------------------------------------------
__global__ __launch_bounds__(256) void edge_exp(
    const int* __restrict__ dst, float* __restrict__ logit_ex,
    const unsigned* __restrict__ mkeys, float* __restrict__ ssum)
{
    const int e = (int)(blockIdx.x * blockDim.x + threadIdx.x);
    if (e >= E2) return;
    const int d = (e < N_EDGES) ? dst[e] : e - N_EDGES;
    const float m  = dec_f32(mkeys[d]);
    const float ex = __expf(logit_ex[e] - m);
    logit_ex[e] = ex;
    atomicAdd(&ssum[d], ex);
}

// ---------------------------------------------------------------------------
// Pass 3: out[dst] += h[src] * (ex / (ssum[dst]+1e-16)).  One wave / edge,
// float4 per lane, 4 f32 atomics per lane (L2-resident working set).
// ---------------------------------------------------------------------------
__global__ __launch_bounds__(256) void aggregate(
    const int* __restrict__ src, const int* __restrict__ dst,
    const float* __restrict__ ex, const float* __restrict__ ssum,
    const float* __restrict__ H, float* __restrict__ out)
{
    const int wid  = (int)((blockIdx.x * blockDim.x + threadIdx.x) >> 5);
    const int lane = threadIdx.x & 31;
    if (wid >= E2) return;
    int s, d;
    if (wid < N_EDGES) { s = src[wid]; d = dst[wid]; }
    else               { s = d = wid - N_EDGES; }
    const float w = ex[wid] / (ssum[d] + 1e-16f);
    float4 h = ((const float4*)(H + (size_t)s * HID))[lane];
    float* o = out + (size_t)d * HID + lane * 4;
    atomicAdd(o + 0, h.x * w);
    atomicAdd(o + 1, h.y * w);
    atomicAdd(o + 2, h.z * w);
    atomicAdd(o + 3, h.w * w);
}

// ---------------------------------------------------------------------------
// h += bias (per channel); optional relu.
// ---------------------------------------------------------------------------
__global__ __launch_bounds__(256) void bias_act(
    float* __restrict__ h, const float* __restrict__ b, int do_relu)
{
    const size_t i = (size_t)blockIdx.x * blockDim.x + threadIdx.x;
    if (i >= (size_t)N_NODES * HID) return;
    float v = h[i] + b[i & (HID - 1)];
    h[i] = do_relu ? fmaxf(v, 0.f) : v;
}

// ---------------------------------------------------------------------------
// out[n] = relu(dot(h[n], Wl) + bl).  One wave / node.
// ---------------------------------------------------------------------------
__global__ __launch_bounds__(256) void final_linear(
    const float* __restrict__ h, const float* __restrict__ Wl,
    const float* __restrict__ bl, float* __restrict__ out)
{
    const int wid  = (int)((blockIdx.x * blockDim.x + threadIdx.x) >> 5);
    const int lane = threadIdx.x & 31;
    if (wid >= N_NODES) return;
    float4 hv = ((const float4*)(h + (size_t)wid * HID))[lane];
    float4 wv = ((const float4*)Wl)[lane];
    float p = hv.x*wv.x + hv.y*wv.y + hv.z*wv.z + hv.w*wv.w;
#pragma unroll
    for (int off = 16; off; off >>= 1) p += __shfl_xor(p, off, 32);
    if (lane == 0) out[wid] = fmaxf(p + bl[0], 0.f);
}

// ---------------------------------------------------------------------------
extern "C" void kernel_launch(void* const* d_in, const int* in_sizes, int n_in,
                              void* d_out, int out_size, void* d_ws, size_t ws_size,
                              hipStream_t stream)
{
    const float* x    = (const float*)d_in[0];
    const int*   src  = (const int*)  d_in[1];
    const int*   dst  = (const int*)  d_in[2];
    const float* ea   = (const float*)d_in[3];
    const float* W1   = (const float*)d_in[4];
    const float* We1  = (const float*)d_in[5];
    const float* as1  = (const float*)d_in[6];
    const float* ad1  = (const float*)d_in[7];
    const float* ae1  = (const float*)d_in[8];
    const float* b1   = (const float*)d_in[9];
    const float* W2   = (const float*)d_in[10];
    const float* We2  = (const float*)d_in[11];
    const float* as2  = (const float*)d_in[12];
    const float* ad2  = (const float*)d_in[13];
    const float* ae2  = (const float*)d_in[14];
    const float* b2   = (const float*)d_in[15];
    const float* Wl   = (const float*)d_in[16];
    const float* bl   = (const float*)d_in[17];
    float* out = (float*)d_out;

    // workspace carve-up (256B aligned regions)
    char* ws = (char*)d_ws;
    size_t off = 0;
    auto carve = [&](size_t bytes) -> void* {
        void* p = ws + off;
        off = (off + bytes + 255) & ~(size_t)255;
        return p;
    };
    float*    bufA  = (float*)   carve((size_t)N_NODES * HID * sizeof(float));
    float*    bufB  = (float*)   carve((size_t)N_NODES * HID * sizeof(float));
    float*    logit = (float*)   carve((size_t)E2 * sizeof(float));
    float*    alpha = (float*)   carve((size_t)N_NODES * sizeof(float));
    float*    beta  = (float*)   carve((size_t)N_NODES * sizeof(float));
    unsigned* mkeys = (unsigned*)carve((size_t)N_NODES * sizeof(unsigned));
    float*    ssum  = (float*)   carve((size_t)N_NODES * sizeof(float));
    float*    msum  = (float*)   carve(EDIM * sizeof(float));
    float*    weae  = (float*)   carve(EDIM * sizeof(float));
    float*    sloop = (float*)   carve(sizeof(float));

    const int TB = 256;
    const int g_gemm  = N_NODES / 16;                       // 3125 (exact)
    const int g_node  = (N_NODES * 32 + TB - 1) / TB;       // 6250 (exact)
    const int g_edge  = (E2 + TB - 1) / TB;                 // 3321
    const int g_agg   = (E2 * 32 + TB - 1) / TB;            // 106250 (exact)
    const int g_bias  = (N_NODES * HID + TB - 1) / TB;      // 25000

    // edge-attr mean (shared by both layers)
    zero32<<<1, 32, 0, stream>>>(msum);
    mean_ea_sum<<<512, TB, 0, stream>>>(ea, msum);

    // ------------------------- layer 1 -------------------------
    gemm_wmma_f32<<<g_gemm, TB, 0, stream>>>(x, W1, bufA, F_INDIM);
    node_prep<<<g_node, TB, 0, stream>>>(bufA, as1, ad1, alpha, beta, mkeys, ssum, bufB);
    prep_edge_proj<<<1, 32, 0, stream>>>(We1, ae1, msum, weae, sloop);
    edge_logit<<<g_edge, TB, 0, stream>>>(src, dst, ea, alpha, beta, weae, sloop, logit, mkeys);
    edge_exp<<<g_edge, TB, 0, stream>>>(dst, logit, mkeys, ssum);
    aggregate<<<g_agg, TB, 0, stream>>>(src, dst, logit, ssum, bufA, bufB);
    bias_act<<<g_bias, TB, 0, stream>>>(bufB, b1, 1);        // + relu

    // ------------------------- layer 2 -------------------------
    gemm_wmma_f32<<<g_gemm, TB, 0, stream>>>(bufB, W2, bufA, HID);
    node_prep<<<g_node, TB, 0, stream>>>(bufA, as2, ad2, alpha, beta, mkeys, ssum, bufB);
    prep_edge_proj<<<1, 32, 0, stream>>>(We2, ae2, msum, weae, sloop);
    edge_logit<<<g_edge, TB, 0, stream>>>(src, dst, ea, alpha, beta, weae, sloop, logit, mkeys);
    edge_exp<<<g_edge, TB, 0, stream>>>(dst, logit, mkeys, ssum);
    aggregate<<<g_agg, TB, 0, stream>>>(src, dst, logit, ssum, bufA, bufB);
    bias_act<<<g_bias, TB, 0, stream>>>(bufB, b2, 0);        // no relu

    // ------------------------- head -------------------------
    final_linear<<<g_node, TB, 0, stream>>>(bufB, Wl, bl, out);
}